// PointTransformerLayer_20255065768446
// MI455X (gfx1250) — compile-verified
//
#include <hip/hip_runtime.h>
#include <hip/hip_bf16.h>

// Point Transformer layer, MI455X (gfx1250, wave32).
// Kernel A: q/k/v projections via V_WMMA_F32_16X16X4_F32 (fp32 tensor path).
// Kernel B: fused gather + pos-MLP + attention-weight GEMMs (WMMA) + softmax + aggregation.

#define NPTS 40000
#define CCH  128
#define KNB  16
#define EPSV 1e-5f

typedef __attribute__((ext_vector_type(2))) float v2f;
typedef __attribute__((ext_vector_type(8))) float v8f;

// ---------------------------------------------------------------------------
// Kernel A: qkv = x @ {Wq,Wk,Wv} + b   (N x 128) @ (128 x 128)
// grid = (N/16, 3), block = 256 (8 waves). Wave w computes cols [16w,16w+16).
// ---------------------------------------------------------------------------
__global__ void __launch_bounds__(256)
qkv_proj_wmma(const float* __restrict__ x,
              const float* __restrict__ Wq, const float* __restrict__ bq,
              const float* __restrict__ Wk, const float* __restrict__ bk,
              const float* __restrict__ Wv, const float* __restrict__ bv,
              float* __restrict__ qp, float* __restrict__ kp, float* __restrict__ vp) {
    __shared__ float xa[16 * CCH];   // 16-row panel of x, 8 KB

    const int m0 = blockIdx.x * 16;
    const float* W; const float* bias; float* outp;
    if (blockIdx.y == 0)      { W = Wq; bias = bq; outp = qp; }
    else if (blockIdx.y == 1) { W = Wk; bias = bk; outp = kp; }
    else                      { W = Wv; bias = bv; outp = vp; }

    const int t = threadIdx.x;
    for (int i = t; i < 16 * CCH; i += 256) xa[i] = x[m0 * CCH + i];
    __syncthreads();

    const int wave = t >> 5;
    const int lane = t & 31;
    const int j0   = wave * 16;            // output column tile
    const int mm   = lane & 15;            // A row (M) owned by this lane
    const int koff = (lane < 16) ? 0 : 2;  // K sub-offset per ISA A/B striping
    const int col  = lane & 15;            // B/D column owned by this lane

    v8f acc = {};
    for (int k0 = 0; k0 < CCH; k0 += 4) {
        v2f a, b;
        // A 16x4 fp32: v0 = {K=k0 | K=k0+2}, v1 = {K=k0+1 | K=k0+3}
        a.x = xa[mm * CCH + k0 + koff];
        a.y = xa[mm * CCH + k0 + koff + 1];
        // B 4x16 fp32: rows of W, striped the same way across lane halves
        b.x = W[(k0 + koff) * CCH + j0 + col];
        b.y = W[(k0 + koff + 1) * CCH + j0 + col];
        acc = __builtin_amdgcn_wmma_f32_16x16x4_f32(false, a, false, b,
                                                    (short)0, acc, false, false);
    }

    const float bb = bias[j0 + col];
    const int rowadd = (lane < 16) ? 0 : 8;   // D: vgpr r holds M=r (lo lanes) / M=r+8 (hi)
    for (int r = 0; r < 8; ++r)
        outp[(size_t)(m0 + r + rowadd) * CCH + j0 + col] = acc[r] + bb;
}

// ---------------------------------------------------------------------------
// Kernel B: fused neighbor attention. 4 points / block, 128 threads (4 waves).
// Rows r = nl*16 + k for nl in [0,4), k in [0,16).
// ---------------------------------------------------------------------------
__global__ void __launch_bounds__(128)
fused_attn(const float* __restrict__ xyz, const float* __restrict__ feats,
           const int* __restrict__ nei,
           const float* __restrict__ Wp1, const float* __restrict__ bp1,
           const float* __restrict__ p_gamma, const float* __restrict__ p_beta,
           const float* __restrict__ p_mean,  const float* __restrict__ p_var,
           const float* __restrict__ Wp2, const float* __restrict__ bp2,
           const float* __restrict__ w_gamma, const float* __restrict__ w_beta,
           const float* __restrict__ w_mean,  const float* __restrict__ w_var,
           const float* __restrict__ Ww1, const float* __restrict__ bw1,
           const float* __restrict__ w1_gamma, const float* __restrict__ w1_beta,
           const float* __restrict__ w1_mean,  const float* __restrict__ w1_var,
           const float* __restrict__ Ww2, const float* __restrict__ bw2,
           const float* __restrict__ qp, const float* __restrict__ kp,
           const float* __restrict__ vp, float* __restrict__ out) {
    __shared__ float w_buf[64 * CCH];   // relu(bn_w(kf - q + p)), 32 KB
    __shared__ float mid[64 * 16];      // 128->16 stage, then softmaxed weights
    __shared__ float qs[4 * CCH];       // q rows for the 4 points
    __shared__ float hbuf[64 * 3];      // relu(bn(d @ Wp1)) per (n,k)
    __shared__ float posn[4 * 3];
    __shared__ int   idxb[64];

    const int t  = threadIdx.x;        // 0..127
    const int n0 = blockIdx.x * 4;

    // ---- setup: q rows, positions, neighbor ids, 3-d positional hidden vec
    for (int i = t; i < 4 * CCH; i += 128) qs[i] = qp[(size_t)n0 * CCH + i];
    if (t < 12) posn[t] = xyz[n0 * 3 + t];
    __syncthreads();

    if (t < 64) {
        const int nl = t >> 4, k = t & 15;
        const int id = nei[(size_t)(n0 + nl) * KNB + k];
        idxb[t] = id;
        const float d0 = xyz[(size_t)id * 3 + 0] - posn[nl * 3 + 0];
        const float d1 = xyz[(size_t)id * 3 + 1] - posn[nl * 3 + 1];
        const float d2 = xyz[(size_t)id * 3 + 2] - posn[nl * 3 + 2];
        for (int j = 0; j < 3; ++j) {
            float h = d0 * Wp1[0 * 3 + j] + d1 * Wp1[1 * 3 + j] + d2 * Wp1[2 * 3 + j] + bp1[j];
            h = (h - p_mean[j]) * (p_gamma[j] * rsqrtf(p_var[j] + EPSV)) + p_beta[j];
            hbuf[t * 3 + j] = fmaxf(h, 0.0f);
        }
    }
    __syncthreads();

    // ---- phase 1: w_buf[r][c] = relu(bn_w(kf - q + p)); gathers are coalesced 512B rows
    const int tc = t;                              // channel owned by this thread
    const float pw0 = Wp2[0 * CCH + tc], pw1 = Wp2[1 * CCH + tc],
                pw2 = Wp2[2 * CCH + tc], bpc = bp2[tc];
    const float wg = w_gamma[tc] * rsqrtf(w_var[tc] + EPSV);
    const float wm = w_mean[tc], wbt = w_beta[tc];
    for (int r = 0; r < 64; ++r) {
        const int id = idxb[r];
        const float kf = kp[(size_t)id * CCH + tc];
        const float pv = hbuf[r * 3 + 0] * pw0 + hbuf[r * 3 + 1] * pw1 +
                         hbuf[r * 3 + 2] * pw2 + bpc;
        float wv = kf - qs[(r >> 4) * CCH + tc] + pv;
        wv = (wv - wm) * wg + wbt;
        w_buf[r * CCH + tc] = fmaxf(wv, 0.0f);
    }
    __syncthreads();

    // ---- phase 2: (64x128)@(128x16) via WMMA, one 16x16 tile per wave (tile = point)
    const int wave = t >> 5;
    const int lane = t & 31;
    const int m0r  = wave * 16;
    const int mm   = lane & 15;
    const int koff = (lane < 16) ? 0 : 2;
    const int col  = lane & 15;

    v8f acc = {};
    for (int k0 = 0; k0 < CCH; k0 += 4) {
        v2f a, b;
        a.x = w_buf[(m0r + mm) * CCH + k0 + koff];
        a.y = w_buf[(m0r + mm) * CCH + k0 + koff + 1];
        b.x = Ww1[(k0 + koff) * 16 + col];
        b.y = Ww1[(k0 + koff + 1) * 16 + col];
        acc = __builtin_amdgcn_wmma_f32_16x16x4_f32(false, a, false, b,
                                                    (short)0, acc, false, false);
    }
    // bias + BN1 + ReLU (per output column), scatter to LDS in row-major
    {
        const float g1 = w1_gamma[col] * rsqrtf(w1_var[col] + EPSV);
        const float m1 = w1_mean[col], b1 = w1_beta[col], bb1 = bw1[col];
        const int rowadd = (lane < 16) ? 0 : 8;
        for (int r = 0; r < 8; ++r) {
            const float v = (acc[r] + bb1 - m1) * g1 + b1;
            mid[(m0r + r + rowadd) * 16 + col] = fmaxf(v, 0.0f);
        }
    }
    __syncthreads();

    // ---- phase 3: (64x16)@(16x16) WMMA, then softmax over the 16 neighbors of the tile
    v8f acc2 = {};
    for (int k0 = 0; k0 < 16; k0 += 4) {
        v2f a, b;
        a.x = mid[(m0r + mm) * 16 + k0 + koff];
        a.y = mid[(m0r + mm) * 16 + k0 + koff + 1];
        b.x = Ww2[(k0 + koff) * 16 + col];
        b.y = Ww2[(k0 + koff + 1) * 16 + col];
        acc2 = __builtin_amdgcn_wmma_f32_16x16x4_f32(false, a, false, b,
                                                     (short)0, acc2, false, false);
    }
    {
        const float bb2 = bw2[col];
        float vals[8];
        float mx = -3.4e38f;
        for (int r = 0; r < 8; ++r) { vals[r] = acc2[r] + bb2; mx = fmaxf(mx, vals[r]); }
        mx = fmaxf(mx, __shfl_xor(mx, 16, 32));     // lanes l / l+16 hold rows r / r+8
        float sum = 0.0f;
        for (int r = 0; r < 8; ++r) { vals[r] = __expf(vals[r] - mx); sum += vals[r]; }
        sum += __shfl_xor(sum, 16, 32);
        const float inv = 1.0f / sum;
        const int rowadd = (lane < 16) ? 0 : 8;
        for (int r = 0; r < 8; ++r)
            mid[(m0r + r + rowadd) * 16 + col] = vals[r] * inv;
    }
    __syncthreads();

    // ---- phase 4: grouped weighted aggregation + residual + leaky ReLU
    // channel c -> weight column j = c % 16 (reshape (S=8, C/S=16))
    const int jw = tc & 15;
    for (int nl = 0; nl < 4; ++nl) {
        float s = 0.0f;
        for (int k = 0; k < KNB; ++k) {
            const int r  = nl * 16 + k;
            const int id = idxb[r];
            const float vf = vp[(size_t)id * CCH + tc];
            const float pv = hbuf[r * 3 + 0] * pw0 + hbuf[r * 3 + 1] * pw1 +
                             hbuf[r * 3 + 2] * pw2 + bpc;
            s += (vf + pv) * mid[r * 16 + jw];
        }
        const float o = s + feats[(size_t)(n0 + nl) * CCH + tc];
        out[(size_t)(n0 + nl) * CCH + tc] = (o > 0.0f) ? o : 0.1f * o;
    }
}

// ---------------------------------------------------------------------------
extern "C" void kernel_launch(void* const* d_in, const int* in_sizes, int n_in,
                              void* d_out, int out_size, void* d_ws, size_t ws_size,
                              hipStream_t stream) {
    const float* xyz      = (const float*)d_in[0];
    const float* feats    = (const float*)d_in[1];
    const int*   nei      = (const int*)  d_in[2];
    const float* Wq       = (const float*)d_in[3];
    const float* bq       = (const float*)d_in[4];
    const float* Wk       = (const float*)d_in[5];
    const float* bk       = (const float*)d_in[6];
    const float* Wv       = (const float*)d_in[7];
    const float* bv       = (const float*)d_in[8];
    const float* Wp1      = (const float*)d_in[9];
    const float* bp1      = (const float*)d_in[10];
    const float* p_gamma  = (const float*)d_in[11];
    const float* p_beta   = (const float*)d_in[12];
    const float* p_mean   = (const float*)d_in[13];
    const float* p_var    = (const float*)d_in[14];
    const float* Wp2      = (const float*)d_in[15];
    const float* bp2      = (const float*)d_in[16];
    const float* w_gamma  = (const float*)d_in[17];
    const float* w_beta   = (const float*)d_in[18];
    const float* w_mean   = (const float*)d_in[19];
    const float* w_var    = (const float*)d_in[20];
    const float* Ww1      = (const float*)d_in[21];
    const float* bw1      = (const float*)d_in[22];
    const float* w1_gamma = (const float*)d_in[23];
    const float* w1_beta  = (const float*)d_in[24];
    const float* w1_mean  = (const float*)d_in[25];
    const float* w1_var   = (const float*)d_in[26];
    const float* Ww2      = (const float*)d_in[27];
    const float* bw2      = (const float*)d_in[28];

    // workspace: q, k, v projections (3 * N * C floats = 58.6 MB)
    float* qp = (float*)d_ws;
    float* kp = qp + (size_t)NPTS * CCH;
    float* vp = kp + (size_t)NPTS * CCH;

    dim3 gridA(NPTS / 16, 3);
    qkv_proj_wmma<<<gridA, 256, 0, stream>>>(feats, Wq, bq, Wk, bk, Wv, bv, qp, kp, vp);

    fused_attn<<<NPTS / 4, 128, 0, stream>>>(
        xyz, feats, nei,
        Wp1, bp1, p_gamma, p_beta, p_mean, p_var, Wp2, bp2,
        w_gamma, w_beta, w_mean, w_var,
        Ww1, bw1, w1_gamma, w1_beta, w1_mean, w1_var, Ww2, bw2,
        qp, kp, vp, (float*)d_out);
}